// GNN_8881992368566
// MI455X (gfx1250) — compile-verified
//
#include <hip/hip_runtime.h>

typedef __attribute__((ext_vector_type(2))) float v2f;
typedef __attribute__((ext_vector_type(8))) float v8f;

#define DIN  128
#define DHID 128
#define DOUTW 64

// ---------------- degree / normalization ----------------

__global__ void init_deg_kernel(float* __restrict__ deg, int n) {
    int i = blockIdx.x * blockDim.x + threadIdx.x;
    if (i < n) deg[i] = 1.0f;   // self-loop contributes 1
}

__global__ void deg_accum_kernel(const int* __restrict__ dst, float* __restrict__ deg, int e) {
    int i = blockIdx.x * blockDim.x + threadIdx.x;
    if (i < e) unsafeAtomicAdd(deg + dst[i], 1.0f);
}

__global__ void rsqrt_kernel(float* __restrict__ deg, int n) {
    int i = blockIdx.x * blockDim.x + threadIdx.x;
    if (i < n) deg[i] = rsqrtf(deg[i]);
}

// ---------------- dense GEMM via fp32 WMMA (16x16x4) ----------------
// A: [nrows x 128] row-major (optionally ReLU'd on load), W: [128 x DO] row-major,
// C: [nrows x DO].  One wave computes a 16-row strip.
//
// W is staged in LDS pre-swizzled so each lane's (W[k][n], W[k+1][n]) pair is one
// contiguous 8-byte element -> single ds_load_b64 straight into an aligned VGPR
// pair (no v_mov repacking).  kp-row pitch = 2*DO+32 floats == 32 banks (mod 64),
// so the second half-wave (K+2/K+3) hits the other 32 banks: conflict-free.

template <int DO, bool RELU>
__global__ __launch_bounds__(256) void gemm_wmma_f32(const float* __restrict__ A,
                                                     const float* __restrict__ W,
                                                     float* __restrict__ C,
                                                     int nrows) {
    constexpr int NT    = DO / 16;        // 16-col tiles
    constexpr int PITCH = 2 * DO + 32;    // floats per kp-row (64 kp-rows)
    __shared__ __align__(16) float Wl[64 * PITCH];

    // swizzled fill: pair p = (kp, n) -> Wl[kp*PITCH + 2n] = {W[2kp][n], W[2kp+1][n]}
    for (int t = threadIdx.x; t < 64 * DO; t += 256) {
        int kp = t / DO, n = t % DO;
        float2 v;
        v.x = W[(2 * kp) * DO + n];
        v.y = W[(2 * kp + 1) * DO + n];
        *(float2*)(&Wl[kp * PITCH + 2 * n]) = v;
    }
    __syncthreads();

    const int wave    = threadIdx.x >> 5;
    const int lane    = threadIdx.x & 31;
    const int rowbase = blockIdx.x * 128 + wave * 16;
    if (rowbase >= nrows) return;          // wave-uniform: EXEC stays all-ones

    const int m   = lane & 15;             // M (A row) / N-within-tile (B,C col)
    const int kh  = (lane >> 4) << 1;      // lanes 16-31 hold K+2,K+3
    const int kpo = lane >> 4;             // kp offset for B
    const float* arow = A + (size_t)(rowbase + m) * DIN;

    v8f acc[NT];
#pragma unroll
    for (int nt = 0; nt < NT; ++nt)
        acc[nt] = (v8f){0.f, 0.f, 0.f, 0.f, 0.f, 0.f, 0.f, 0.f};

    for (int k0 = 0; k0 < DIN; k0 += 4) {
        v2f a;
        a.x = arow[k0 + kh];
        a.y = arow[k0 + kh + 1];
        if (RELU) { a.x = fmaxf(a.x, 0.f); a.y = fmaxf(a.y, 0.f); }
        const float* brow = &Wl[((k0 >> 1) + kpo) * PITCH + 2 * m];
#pragma unroll
        for (int nt = 0; nt < NT; ++nt) {
            v2f b = *(const v2f*)(brow + 32 * nt);   // ds_load_b64
            acc[nt] = __builtin_amdgcn_wmma_f32_16x16x4_f32(
                false, a, false, b, (short)0, acc[nt], false, false);
        }
    }

    // C/D layout: VGPR r holds row (r) for lanes 0-15, row (r+8) for lanes 16-31
    const int row2 = rowbase + ((lane >> 4) << 3);
#pragma unroll
    for (int nt = 0; nt < NT; ++nt) {
        const int n = nt * 16 + m;
#pragma unroll
        for (int r = 0; r < 8; ++r)
            C[(size_t)(row2 + r) * DO + n] = acc[nt][r];
    }
}

// ---------------- self-loop + bias init (non-atomic, erases poison) ----------------

template <int D>
__global__ void self_bias_kernel(const float* __restrict__ h,
                                 const float* __restrict__ inv,
                                 const float* __restrict__ bias,
                                 float* __restrict__ agg, int n) {
    long t = (long)blockIdx.x * blockDim.x + threadIdx.x;
    if (t >= (long)n * (D / 4)) return;
    int i  = (int)(t / (D / 4));
    int j4 = ((int)(t % (D / 4))) * 4;
    float c = inv[i]; c *= c;                  // self-loop coef = inv[i]^2
    float4 v  = *(const float4*)(h + (size_t)i * D + j4);
    float4 bb = *(const float4*)(bias + j4);
    float4 o;
    o.x = v.x * c + bb.x; o.y = v.y * c + bb.y;
    o.z = v.z * c + bb.z; o.w = v.w * c + bb.w;
    *(float4*)(agg + (size_t)i * D + j4) = o;
}

// ---------------- edge scatter: gather h[src]*coef, atomic-add into agg[dst] ----------------
// D=128: one wave per edge (32 lanes x float4).  D=64: half-wave per edge.
// Gathers and atomics resolve in the 192MB L2 (h and agg both fit).

template <int D>
__global__ void edge_scatter_kernel(const int* __restrict__ src,
                                    const int* __restrict__ dst,
                                    const float* __restrict__ inv,
                                    const float* __restrict__ h,
                                    float* __restrict__ agg, int e) {
    const int LPE = D / 4;                      // lanes per edge
    long t = (long)blockIdx.x * blockDim.x + threadIdx.x;
    long ed = t / LPE;
    if (ed >= e) return;
    int lane = (int)(t % LPE);
    int s = src[ed], d = dst[ed];
    float c = inv[s] * inv[d];
    float4 v = *(const float4*)(h + (size_t)s * D + lane * 4);
    float* p = agg + (size_t)d * D + lane * 4;
    unsafeAtomicAdd(p + 0, v.x * c);
    unsafeAtomicAdd(p + 1, v.y * c);
    unsafeAtomicAdd(p + 2, v.z * c);
    unsafeAtomicAdd(p + 3, v.w * c);
}

// ---------------- launch ----------------

extern "C" void kernel_launch(void* const* d_in, const int* in_sizes, int n_in,
                              void* d_out, int out_size, void* d_ws, size_t ws_size,
                              hipStream_t stream) {
    (void)n_in; (void)out_size; (void)ws_size;
    const float* x  = (const float*)d_in[0];
    const int*   ei = (const int*)d_in[1];
    const float* W1 = (const float*)d_in[2];
    const float* b1 = (const float*)d_in[3];
    const float* W2 = (const float*)d_in[4];
    const float* b2 = (const float*)d_in[5];
    float* out = (float*)d_out;

    const int N = in_sizes[0] / DIN;     // 100000
    const int E = in_sizes[1] / 2;       // 1600000
    const int* src = ei;                 // edge_index[0]
    const int* dst = ei + E;             // edge_index[1]

    float* deg = (float*)d_ws;                                   // N (holds inv_sqrt after pass)
    float* hx  = deg + (((size_t)N + 255) & ~(size_t)255);       // N*128 (x@W1; reused as h2)
    float* ag1 = hx + (size_t)N * DHID;                          // N*128 (layer-1 accumulator)
    float* h2  = hx;                                             // N*64 reuses hx

    const int TB = 256;

    // normalization coefficients
    init_deg_kernel<<<(N + TB - 1) / TB, TB, 0, stream>>>(deg, N);
    deg_accum_kernel<<<(E + TB - 1) / TB, TB, 0, stream>>>(dst, deg, E);
    rsqrt_kernel<<<(N + TB - 1) / TB, TB, 0, stream>>>(deg, N);

    // layer 1: hx = x @ W1  (fp32 WMMA)
    gemm_wmma_f32<DHID, false><<<(N + 127) / 128, TB, 0, stream>>>(x, W1, hx, N);
    {   // agg1 = self-loop + bias, then edge scatter (atomics)
        long tot = (long)N * (DHID / 4);
        self_bias_kernel<DHID><<<(unsigned)((tot + TB - 1) / TB), TB, 0, stream>>>(hx, deg, b1, ag1, N);
    }
    {
        long tot = (long)E * (DHID / 4);
        edge_scatter_kernel<DHID><<<(unsigned)((tot + TB - 1) / TB), TB, 0, stream>>>(src, dst, deg, hx, ag1, E);
    }

    // layer 2: h2 = relu(agg1) @ W2  (ReLU fused into the A load), output into d_out
    gemm_wmma_f32<DOUTW, true><<<(N + 127) / 128, TB, 0, stream>>>(ag1, W2, h2, N);
    {
        long tot = (long)N * (DOUTW / 4);
        self_bias_kernel<DOUTW><<<(unsigned)((tot + TB - 1) / TB), TB, 0, stream>>>(h2, deg, b2, out, N);
    }
    {
        long tot = (long)E * (DOUTW / 4);
        edge_scatter_kernel<DOUTW><<<(unsigned)((tot + TB - 1) / TB), TB, 0, stream>>>(src, dst, deg, h2, out, E);
    }
}